// Correlation3D_74552042324063
// MI455X (gfx1250) — compile-verified
//
#include <hip/hip_runtime.h>

#define B_ 2
#define N_ 8192
#define C_ 64
#define O_ 64
#define K_ 16
#define CIN 131           // 3 + 2*C
#define SLOPE 0.1f

typedef float v2f __attribute__((ext_vector_type(2)));
typedef float v8f __attribute__((ext_vector_type(8)));

__device__ __forceinline__ float lrelu_(float x) { return x >= 0.f ? x : SLOPE * x; }
__device__ __forceinline__ float relu_(float x)  { return x > 0.f ? x : 0.f; }

// ---------------------------------------------------------------------------
// Kernel 1: brute-force KNN (K=16) with LDS-tiled candidates.
// One thread per query point; top-16 kept in registers via unrolled swap chain.
// ---------------------------------------------------------------------------
__global__ __launch_bounds__(256) void knn_kernel(const float* __restrict__ q,
                                                  const float* __restrict__ in,
                                                  int* __restrict__ out_idx) {
  __shared__ float sx[256], sy[256], sz[256];
  int tid = blockIdx.x * 256 + threadIdx.x;
  int b = tid / N_;
  int n = tid % N_;
  const float* qb = q + (size_t)b * 3 * N_;
  const float* ib = in + (size_t)b * 3 * N_;
  float qx = qb[n], qy = qb[N_ + n], qz = qb[2 * N_ + n];

  float bd[K_];
  int   bi[K_];
#pragma unroll
  for (int i = 0; i < K_; ++i) { bd[i] = 3.4e38f; bi[i] = 0; }

  for (int tile = 0; tile < N_; tile += 256) {
    int j = tile + threadIdx.x;
    sx[threadIdx.x] = ib[j];
    sy[threadIdx.x] = ib[N_ + j];
    sz[threadIdx.x] = ib[2 * N_ + j];
    __syncthreads();
    for (int t = 0; t < 256; ++t) {
      float dx = qx - sx[t], dy = qy - sy[t], dz = qz - sz[t];
      float d = dx * dx + dy * dy + dz * dz;
      if (d < bd[K_ - 1]) {             // rare after warm-up
        float cd = d; int ci = tile + t;
#pragma unroll
        for (int u = 0; u < K_; ++u) {  // sorted-insert via swap chain (regs only)
          if (cd < bd[u]) {
            float tf = bd[u]; bd[u] = cd; cd = tf;
            int   ti = bi[u]; bi[u] = ci; ci = ti;
          }
        }
      }
    }
    __syncthreads();
  }
#pragma unroll
  for (int u = 0; u < K_; ++u) out_idx[(size_t)(b * N_ + n) * K_ + u] = bi[u];
}

// ---------------------------------------------------------------------------
// Kernel 2: cost_mlp (131->64->64, leaky-relu) via fp32 WMMA 16x16x4,
// weight_net2 (3->8->8->64, relu), and reduction over k -> p2n_cost[B,O,N].
// One wave per query point: 16 neighbor-rows x 64 outputs tile.
// Layer-0 K-loop is split into three branch-free phases on the exact
// 4-aligned channel boundaries: feat1 [0,64) | feat2 [64,128) | xyz+pad.
// ---------------------------------------------------------------------------
__global__ __launch_bounds__(256) void cost_kernel(
    const float* __restrict__ xyz1, const float* __restrict__ feat1,
    const float* __restrict__ xyz2, const float* __restrict__ feat2,
    const float* __restrict__ cm_w0, const float* __restrict__ cm_b0,
    const float* __restrict__ cm_w1, const float* __restrict__ cm_b1,
    const float* __restrict__ w2_w0, const float* __restrict__ w2_b0,
    const float* __restrict__ w2_w1, const float* __restrict__ w2_b1,
    const float* __restrict__ w2_w2, const float* __restrict__ w2_b2,
    const int* __restrict__ idx12, float* __restrict__ p2n) {
  __shared__ float Hbuf[8][16][65];   // per-wave 16x64 tile (+1 pad vs 64 banks)
  __shared__ float h2buf[8][16][8];   // per-wave weight_net2 hidden

  int wave = threadIdx.x >> 5, lane = threadIdx.x & 31;
  int lo = lane & 15, hi = lane >> 4;
  int p = blockIdx.x * 8 + wave;
  int b = p / N_, n = p % N_;

  int m = idx12[(size_t)(b * N_ + n) * K_ + lo];   // neighbor for row `lo`
  float dx = xyz2[((size_t)b * 3 + 0) * N_ + m] - xyz1[((size_t)b * 3 + 0) * N_ + n];
  float dy = xyz2[((size_t)b * 3 + 1) * N_ + m] - xyz1[((size_t)b * 3 + 1) * N_ + n];
  float dz = xyz2[((size_t)b * 3 + 2) * N_ + m] - xyz1[((size_t)b * 3 + 2) * N_ + n];

  // hoisted per-lane bases: A rows and B weight rows
  const float* f1b = feat1 + ((size_t)b * C_) * N_ + n;   // [ch] stride N_
  const float* f2b = feat2 + ((size_t)b * C_) * N_ + m;   // [ch] stride N_
  const float* w0r[4];
  const float* w1r[4];
#pragma unroll
  for (int t = 0; t < 4; ++t) {
    w0r[t] = cm_w0 + (t * 16 + lo) * CIN;
    w1r[t] = cm_w1 + (t * 16 + lo) * 64;
  }

  const v8f vzero = {0.f, 0.f, 0.f, 0.f, 0.f, 0.f, 0.f, 0.f};
  v8f acc[4] = {vzero, vzero, vzero, vzero};

  // ---- layer 0, phase 1: feat1 channels [0,64) — branch-free ----
#pragma unroll 4
  for (int k0 = 0; k0 < 64; k0 += 4) {
    int kk = k0 + 2 * hi;   // A: low half K=k0,k0+1; high half K=k0+2,k0+3
    v2f A;
    A.x = f1b[(size_t)kk * N_];
    A.y = f1b[(size_t)(kk + 1) * N_];
#pragma unroll
    for (int t = 0; t < 4; ++t) {
      v2f Bv;
      Bv.x = w0r[t][kk];
      Bv.y = w0r[t][kk + 1];
      acc[t] = __builtin_amdgcn_wmma_f32_16x16x4_f32(false, A, false, Bv,
                                                     (short)0, acc[t], false, false);
    }
  }
  // ---- layer 0, phase 2: feat2 channels [64,128) — branch-free gather ----
#pragma unroll 4
  for (int k0 = 0; k0 < 64; k0 += 4) {
    int kk = k0 + 2 * hi;
    v2f A;
    A.x = f2b[(size_t)kk * N_];
    A.y = f2b[(size_t)(kk + 1) * N_];
#pragma unroll
    for (int t = 0; t < 4; ++t) {
      v2f Bv;
      Bv.x = w0r[t][64 + kk];
      Bv.y = w0r[t][64 + kk + 1];
      acc[t] = __builtin_amdgcn_wmma_f32_16x16x4_f32(false, A, false, Bv,
                                                     (short)0, acc[t], false, false);
    }
  }
  // ---- layer 0, phase 3: xyz channels 128..130 (+ zero pad 131) ----
  {
    v2f A;
    A.x = hi ? dz : dx;           // low half ch 128,129 ; high half ch 130,131
    A.y = hi ? 0.f : dy;
#pragma unroll
    for (int t = 0; t < 4; ++t) {
      v2f Bv;
      Bv.x = w0r[t][hi ? 130 : 128];
      Bv.y = hi ? 0.f : w0r[t][129];
      acc[t] = __builtin_amdgcn_wmma_f32_16x16x4_f32(false, A, false, Bv,
                                                     (short)0, acc[t], false, false);
    }
  }

  // bias + leaky-relu, stage hidden to LDS in row-major [row][o] for layer-1 A
#pragma unroll
  for (int t = 0; t < 4; ++t) {
    float bias = cm_b0[t * 16 + lo];
#pragma unroll
    for (int j = 0; j < 8; ++j)         // C layout: row = j + 8*hi, col = t*16+lo
      Hbuf[wave][j + 8 * hi][t * 16 + lo] = lrelu_(acc[t][j] + bias);
  }
  __syncthreads();

  // ---- layer 1: [16 x 64] x [64 x 64], branch-free ----
  v8f acc2[4] = {vzero, vzero, vzero, vzero};
#pragma unroll 4
  for (int k0 = 0; k0 < 64; k0 += 4) {
    int kk = k0 + 2 * hi;
    v2f A;
    A.x = Hbuf[wave][lo][kk];
    A.y = Hbuf[wave][lo][kk + 1];
#pragma unroll
    for (int t = 0; t < 4; ++t) {
      v2f Bv;
      Bv.x = w1r[t][kk];
      Bv.y = w1r[t][kk + 1];
      acc2[t] = __builtin_amdgcn_wmma_f32_16x16x4_f32(false, A, false, Bv,
                                                      (short)0, acc2[t], false, false);
    }
  }

  // ---- weight_net2 on (dx,dy,dz): 3 -> 8 -> 8, relu ----
  float h1[8];
#pragma unroll
  for (int i = 0; i < 8; ++i)
    h1[i] = relu_(w2_w0[i * 3 + 0] * dx + w2_w0[i * 3 + 1] * dy +
                  w2_w0[i * 3 + 2] * dz + w2_b0[i]);
  float h2r[8];
#pragma unroll
  for (int i = 0; i < 8; ++i) {
    float s = w2_b1[i];
#pragma unroll
    for (int j = 0; j < 8; ++j) s += w2_w1[i * 8 + j] * h1[j];
    h2r[i] = relu_(s);
  }
  if (lane < 16) {
#pragma unroll
    for (int j = 0; j < 8; ++j) h2buf[wave][lo][j] = h2r[j];
  }
  __syncthreads();

  // ---- last layer of weight_net2 (8 -> 64), written into reused Hbuf ----
  for (int i = 0; i < 32; ++i) {
    int e = i * 32 + lane;
    int o = e & 63, row = e >> 6;
    float s = w2_b2[o];
#pragma unroll
    for (int j = 0; j < 8; ++j) s += w2_w2[o * 8 + j] * h2buf[wave][row][j];
    Hbuf[wave][row][o] = relu_(s);
  }
  __syncthreads();

  // ---- p2n_cost[o][n] = sum_k weights2[k][o] * lrelu(p2p[k][o] + b1[o]) ----
  float part[4];
#pragma unroll
  for (int t = 0; t < 4; ++t) {
    float bias = cm_b1[t * 16 + lo];
    float s = 0.f;
#pragma unroll
    for (int j = 0; j < 8; ++j)
      s += lrelu_(acc2[t][j] + bias) * Hbuf[wave][j + 8 * hi][t * 16 + lo];
    s += __shfl_xor(s, 16, 32);   // fold the other 8 rows (opposite half-wave)
    part[t] = s;
  }
  if (lane < 16) {
#pragma unroll
    for (int t = 0; t < 4; ++t)
      p2n[((size_t)b * O_ + t * 16 + lane) * N_ + n] = part[t];
  }
}

// ---------------------------------------------------------------------------
// Kernel 3: weight_net1 on self-KNN deltas + gather p2n_cost and reduce over k.
// One wave per query point; each lane owns outputs {lane, lane+32}.
// ---------------------------------------------------------------------------
__global__ __launch_bounds__(256) void final_kernel(
    const float* __restrict__ xyz1,
    const float* __restrict__ w1_w0, const float* __restrict__ w1_b0,
    const float* __restrict__ w1_w1, const float* __restrict__ w1_b1,
    const float* __restrict__ w1_w2, const float* __restrict__ w1_b2,
    const int* __restrict__ idx11, const float* __restrict__ p2n,
    float* __restrict__ out) {
  __shared__ float h2buf[8][16][8];
  __shared__ int idxbuf[8][16];

  int wave = threadIdx.x >> 5, lane = threadIdx.x & 31;
  int lo = lane & 15;
  int p = blockIdx.x * 8 + wave;
  int b = p / N_, n = p % N_;

  int m = idx11[(size_t)(b * N_ + n) * K_ + lo];
  const float* xb = xyz1 + (size_t)b * 3 * N_;
  float dx = xb[m] - xb[n];
  float dy = xb[N_ + m] - xb[N_ + n];
  float dz = xb[2 * N_ + m] - xb[2 * N_ + n];

  float h1[8];
#pragma unroll
  for (int i = 0; i < 8; ++i)
    h1[i] = relu_(w1_w0[i * 3 + 0] * dx + w1_w0[i * 3 + 1] * dy +
                  w1_w0[i * 3 + 2] * dz + w1_b0[i]);
  float h2r[8];
#pragma unroll
  for (int i = 0; i < 8; ++i) {
    float s = w1_b1[i];
#pragma unroll
    for (int j = 0; j < 8; ++j) s += w1_w1[i * 8 + j] * h1[j];
    h2r[i] = relu_(s);
  }
  if (lane < 16) {
    idxbuf[wave][lo] = m;
#pragma unroll
    for (int j = 0; j < 8; ++j) h2buf[wave][lo][j] = h2r[j];
  }
  __syncthreads();

  const float* p2nb = p2n + (size_t)b * O_ * N_;
#pragma unroll
  for (int oi = 0; oi < 2; ++oi) {
    int o = lane + oi * 32;
    float wr[8];
#pragma unroll
    for (int j = 0; j < 8; ++j) wr[j] = w1_w2[o * 8 + j];
    float bo = w1_b2[o];
    float s = 0.f;
    for (int k = 0; k < K_; ++k) {
      float w = bo;
#pragma unroll
      for (int j = 0; j < 8; ++j) w += wr[j] * h2buf[wave][k][j];
      w = relu_(w);
      s += w * p2nb[(size_t)o * N_ + idxbuf[wave][k]];
    }
    out[((size_t)b * O_ + o) * N_ + n] = s;
  }
}

// ---------------------------------------------------------------------------
extern "C" void kernel_launch(void* const* d_in, const int* in_sizes, int n_in,
                              void* d_out, int out_size, void* d_ws, size_t ws_size,
                              hipStream_t stream) {
  (void)in_sizes; (void)n_in; (void)out_size; (void)ws_size;
  const float* xyz1  = (const float*)d_in[0];
  const float* feat1 = (const float*)d_in[1];
  const float* xyz2  = (const float*)d_in[2];
  const float* feat2 = (const float*)d_in[3];
  const float* cm_w0 = (const float*)d_in[4];
  const float* cm_b0 = (const float*)d_in[5];
  const float* cm_w1 = (const float*)d_in[6];
  const float* cm_b1 = (const float*)d_in[7];
  const float* w1_w0 = (const float*)d_in[8];
  const float* w1_b0 = (const float*)d_in[9];
  const float* w1_w1 = (const float*)d_in[10];
  const float* w1_b1 = (const float*)d_in[11];
  const float* w1_w2 = (const float*)d_in[12];
  const float* w1_b2 = (const float*)d_in[13];
  const float* w2_w0 = (const float*)d_in[14];
  const float* w2_b0 = (const float*)d_in[15];
  const float* w2_w1 = (const float*)d_in[16];
  const float* w2_b1 = (const float*)d_in[17];
  const float* w2_w2 = (const float*)d_in[18];
  const float* w2_b2 = (const float*)d_in[19];

  // workspace: idx_1in2 (1MB) | idx_1in1 (1MB) | p2n_cost (4MB)
  int* idx12 = (int*)d_ws;
  int* idx11 = idx12 + (size_t)B_ * N_ * K_;
  float* p2n = (float*)(idx11 + (size_t)B_ * N_ * K_);
  float* out = (float*)d_out;

  dim3 blk(256);
  knn_kernel<<<dim3(B_ * N_ / 256), blk, 0, stream>>>(xyz1, xyz2, idx12);
  knn_kernel<<<dim3(B_ * N_ / 256), blk, 0, stream>>>(xyz1, xyz1, idx11);
  cost_kernel<<<dim3(B_ * N_ / 8), blk, 0, stream>>>(
      xyz1, feat1, xyz2, feat2, cm_w0, cm_b0, cm_w1, cm_b1,
      w2_w0, w2_b0, w2_w1, w2_b1, w2_w2, w2_b2, idx12, p2n);
  final_kernel<<<dim3(B_ * N_ / 8), blk, 0, stream>>>(
      xyz1, w1_w0, w1_b0, w1_w1, w1_b1, w1_w2, w1_b2, idx11, p2n, out);
}